// PositionalEncoding_56564719288370
// MI455X (gfx1250) — compile-verified
//
#include <hip/hip_runtime.h>
#include <hip/hip_bf16.h>
#include <math.h>

// out = X + PE(8192, 4096), fp32. Memory-bound: 256 MiB @ 23.3 TB/s ~ 11.5us.
// Each block: one 1024-column strip x 8 rows. Column frequencies (v_exp_f32)
// hoisted out of the tile loop; X streamed via gfx1250 async global->LDS
// copies (ASYNCcnt, double-buffered); trig via explicit rotation reduction
// + native v_sin_f32 / v_cos_f32 (inputs in rotations).

#define AS1 __attribute__((address_space(1)))
#define AS3 __attribute__((address_space(3)))

typedef int v4i __attribute__((ext_vector_type(4)));

#ifndef __has_builtin
#define __has_builtin(x) 0
#endif

#if __has_builtin(__builtin_amdgcn_s_wait_asynccnt)
#define WAIT_ASYNC(n) __builtin_amdgcn_s_wait_asynccnt(n)
#else
#define WAIT_ASYNC(n) asm volatile("s_wait_asynccnt " #n ::: "memory")
#endif

__device__ __forceinline__ void async_cp_b128(const float* gsrc, void* ldst) {
#if __has_builtin(__builtin_amdgcn_global_load_async_to_lds_b128)
  __builtin_amdgcn_global_load_async_to_lds_b128(
      (AS1 v4i*)(gsrc), (AS3 v4i*)(ldst), /*imm offset*/ 0, /*cpol*/ 0);
#else
  unsigned loff = (unsigned)(unsigned long long)(AS3 void*)ldst;  // LDS byte offset
  unsigned long long ga = (unsigned long long)gsrc;
  asm volatile("global_load_async_to_lds_b128 %0, %1, off"
               :: "v"(loff), "v"(ga) : "memory");
#endif
}

// sin(2*pi*w) with exact periodic reduction into [-0.5, 0.5] rotations.
__device__ __forceinline__ float sin_rot(float w) {
  float r = w - __builtin_rintf(w);
  return __builtin_amdgcn_sinf(r);
}
__device__ __forceinline__ float cos_rot(float w) {
  float r = w - __builtin_rintf(w);
  return __builtin_amdgcn_cosf(r);
}

// 8192 rows x 4096 cols. Block: 256 threads; covers cols [strip*1024, +1024)
// for rows [rowBase, rowBase+8). 4 strips * 1024 row-groups = 4096 blocks.
__global__ __launch_bounds__(256) void pe_add_kernel(const float* __restrict__ X,
                                                     float* __restrict__ out) {
  constexpr int N = 4096;               // row stride (floats)
  constexpr int TILE_FLOATS = 256 * 4;  // 1024 floats per tile (one row chunk)
  constexpr int ROWS = 8;
  __shared__ float lds[2][TILE_FLOATS];

  const unsigned tid = threadIdx.x;
  const unsigned lane4 = tid * 4;
  const unsigned strip = blockIdx.x & 3u;
  const unsigned rowBase = (blockIdx.x >> 2) * ROWS;
  const unsigned colBase = strip * TILE_FLOATS;
  const unsigned base = rowBase * N + colBase;  // float index of tile 0

  // Prime the pipeline: row 0 of this block -> buffer 0.
  async_cp_b128(X + base + lane4, &lds[0][lane4]);

  // Loop-invariant per-lane frequencies, in rotations-per-position:
  //   w_j(pos) = pos * 10000^(-(j + j%2)/n) / (2*pi)
  // Columns j=col..col+3 need exponents col, col+2, col+2, col+4.
  const float s = 13.287712379549449f / 4096.0f;  // log2(10000)/n
  const float inv2pi = 0.15915494309189535f;
  const unsigned col = colBase + lane4;
  const float w0 = exp2f(-(float)(col)*s) * inv2pi;
  const float w2 = exp2f(-(float)(col + 2) * s) * inv2pi;
  const float w4 = exp2f(-(float)(col + 4) * s) * inv2pi;

#pragma unroll
  for (int t = 0; t < ROWS; ++t) {
    if (t + 1 < ROWS) {
      async_cp_b128(X + base + (unsigned)(t + 1) * N + lane4,
                    &lds[(t + 1) & 1][lane4]);
      WAIT_ASYNC(1);  // row t has landed; row t+1 still in flight
    } else {
      WAIT_ASYNC(0);
    }

    const float pos = (float)(rowBase + (unsigned)t);
    const float4 x = *(const float4*)&lds[t & 1][lane4];

    float4 r;
    r.x = x.x + sin_rot(pos * w0);
    r.y = x.y + cos_rot(pos * w2);
    r.z = x.z + sin_rot(pos * w2);
    r.w = x.w + cos_rot(pos * w4);

    *(float4*)(out + base + (unsigned)t * N + lane4) = r;
  }
}

extern "C" void kernel_launch(void* const* d_in, const int* in_sizes, int n_in,
                              void* d_out, int out_size, void* d_ws, size_t ws_size,
                              hipStream_t stream) {
  const float* X = (const float*)d_in[0];
  float* out = (float*)d_out;
  // 8192*4096 floats; each block handles 1024 cols x 8 rows = 8192 floats.
  const int total = in_sizes[0];            // 33,554,432
  const int blocks = total / (256 * 4 * 8); // 4096
  pe_add_kernel<<<dim3(blocks), dim3(256), 0, stream>>>(X, out);
}